// SimpleDilatedAttention_8615704396432
// MI455X (gfx1250) — compile-verified
//
#include <hip/hip_runtime.h>

// Dilated "attention over heads" for (b=4, n=8192, h=8, d=64) fp32.
// Memory-bound (~330MB traffic, ~0.5 GFLOP): 3 streaming pass1 kernels
// (one per head group) write unnormalized x into d_out (zeros at uncovered
// slots) and per-block partial denominator sums (reduced with
// V_WMMA_F32_16X16X4_F32 ones-matrix column-sum trick), then a deterministic
// reduce kernel builds denom[b][h][d], then pass2 rescales d_out in place.

#define BB 4
#define NN 8192
#define HH 8
#define DD 64

typedef __attribute__((ext_vector_type(2))) float v2f;
typedef __attribute__((ext_vector_type(8))) float v8f;

// Workspace layout in floats:
//   [0, 2048)            denom[B][H][D]
//   [WS_P0, +4*256*192)  group0 partials [B][256 blocks][192 cols]
//   [WS_P1, +4*256*192)  group1 partials
//   [WS_P2, +4*256*128)  group2 partials
// total = 526,336 floats (~2.06 MB)
#define WS_DENOM 0
#define WS_P0    2048
#define WS_P1    (2048 + 4 * 256 * 192)
#define WS_P2    (2048 + 2 * 4 * 256 * 192)

// One block = 32 positions x 8 lanes-per-position (each lane owns 8 d's).
template <int G, int HMIN, int RATE, int OFF, int PART_OFF>
__global__ __launch_bounds__(256) void dil_pass1(
    const float* __restrict__ q, const float* __restrict__ k,
    const float* __restrict__ v, float* __restrict__ out,
    float* __restrict__ ws) {
  constexpr int COLS = G * 64;
  __shared__ float xs[32][COLS];

  const int t   = threadIdx.x;
  const int p   = t >> 3;     // position slot within block: 0..31
  const int ld  = t & 7;      // d-chunk: lane handles d in [ld*8, ld*8+8)
  const int b   = blockIdx.y;
  const int pos = blockIdx.x * 32 + p;
  const int base = ((b * NN + pos) * HH + HMIN) * DD + ld * 8;

  float4 xa[G], xb[G];
#pragma unroll
  for (int h = 0; h < G; ++h) {
    xa[h] = make_float4(0.f, 0.f, 0.f, 0.f);
    xb[h] = make_float4(0.f, 0.f, 0.f, 0.f);
  }

  if ((pos & (RATE - 1)) == OFF) {
    float4 qa[G], qb[G], ka[G], kb[G];
#pragma unroll
    for (int h = 0; h < G; ++h) {
      qa[h] = *(const float4*)(q + base + h * DD);
      qb[h] = *(const float4*)(q + base + h * DD + 4);
      ka[h] = *(const float4*)(k + base + h * DD);
      kb[h] = *(const float4*)(k + base + h * DD + 4);
    }
    // scores[qh][kh] = dot_d(q,k): 8-elem partial dot then 8-lane xor reduce
    float s[G][G];
#pragma unroll
    for (int qh = 0; qh < G; ++qh) {
#pragma unroll
      for (int kh = 0; kh < G; ++kh) {
        float a = qa[qh].x * ka[kh].x + qa[qh].y * ka[kh].y +
                  qa[qh].z * ka[kh].z + qa[qh].w * ka[kh].w +
                  qb[qh].x * kb[kh].x + qb[qh].y * kb[kh].y +
                  qb[qh].z * kb[kh].z + qb[qh].w * kb[kh].w;
        a += __shfl_xor(a, 1, 32);
        a += __shfl_xor(a, 2, 32);
        a += __shfl_xor(a, 4, 32);
        s[qh][kh] = a;
      }
    }
    // softmax over kh with scale 1/sqrt(64) = 0.125
    float pr[G][G];
#pragma unroll
    for (int qh = 0; qh < G; ++qh) {
      float m = s[qh][0];
#pragma unroll
      for (int kh = 1; kh < G; ++kh) m = fmaxf(m, s[qh][kh]);
      float sum = 0.f;
#pragma unroll
      for (int kh = 0; kh < G; ++kh) {
        pr[qh][kh] = __expf((s[qh][kh] - m) * 0.125f);
        sum += pr[qh][kh];
      }
      const float inv = 1.f / sum;
#pragma unroll
      for (int kh = 0; kh < G; ++kh) pr[qh][kh] *= inv;
    }
    // x = attn @ V
#pragma unroll
    for (int kh = 0; kh < G; ++kh) {
      const float4 va = *(const float4*)(v + base + kh * DD);
      const float4 vb = *(const float4*)(v + base + kh * DD + 4);
#pragma unroll
      for (int qh = 0; qh < G; ++qh) {
        const float w = pr[qh][kh];
        xa[qh].x += w * va.x; xa[qh].y += w * va.y;
        xa[qh].z += w * va.z; xa[qh].w += w * va.w;
        xb[qh].x += w * vb.x; xb[qh].y += w * vb.y;
        xb[qh].z += w * vb.z; xb[qh].w += w * vb.w;
      }
    }
  }

  // Write unnormalized x to d_out (zeros for uncovered) + stage in LDS
#pragma unroll
  for (int h = 0; h < G; ++h) {
    *(float4*)(out + base + h * DD)     = xa[h];
    *(float4*)(out + base + h * DD + 4) = xb[h];
    *(float4*)(&xs[p][h * 64 + ld * 8])     = xa[h];
    *(float4*)(&xs[p][h * 64 + ld * 8 + 4]) = xb[h];
  }
  __syncthreads();

  // Wave 0: column-sum the 32 x COLS tile with WMMA (ones-matrix trick).
  // D = ones(16x4) x B(4x16) + C -> every row of D is the column sum; chain
  // 8 chunks of 4 position-rows through the accumulator.
  if (t < 32) {
    const v2f A = {1.0f, 1.0f};  // all-ones A: robust to A/B K-layout
    const int n16 = t & 15;
    float* partial = ws + PART_OFF + (b * 256 + blockIdx.x) * COLS;
#pragma unroll
    for (int tile = 0; tile < COLS / 16; ++tile) {
      v8f acc = {};
#pragma unroll
      for (int pc = 0; pc < 8; ++pc) {
        const int r0 = pc * 4 + ((t < 16) ? 0 : 2);
        const v2f Bm = {xs[r0][tile * 16 + n16], xs[r0 + 1][tile * 16 + n16]};
        acc = __builtin_amdgcn_wmma_f32_16x16x4_f32(
            false, A, false, Bm, (short)0, acc, false, false);
      }
      // D row M=0 lives in VGPR0 of lanes 0..15 (N = lane)
      if (t < 16) partial[tile * 16 + t] = acc[0];
    }
  }
}

// Deterministic reduction of per-block partials -> denom[b][h][d]
__global__ __launch_bounds__(256) void dil_reduce(float* __restrict__ ws) {
  const int i = blockIdx.x * 256 + threadIdx.x;
  if (i >= BB * HH * DD) return;
  const int b = i >> 9;
  const int h = (i >> 6) & 7;
  const int d = i & 63;
  const float* part;
  int cols, col;
  if (h < 3)      { part = ws + WS_P0; cols = 192; col = h * 64 + d; }
  else if (h < 6) { part = ws + WS_P1; cols = 192; col = (h - 3) * 64 + d; }
  else            { part = ws + WS_P2; cols = 128; col = (h - 6) * 64 + d; }
  const float* pp = part + (b * 256) * cols + col;
  float sum = 0.f;
  for (int blk = 0; blk < 256; ++blk) sum += pp[blk * cols];
  ws[WS_DENOM + i] = sum;
}

// out = out / (denom + eps) / 3  (uncovered slots are 0 and stay 0)
__global__ __launch_bounds__(256) void dil_pass2(float* __restrict__ out,
                                                 const float* __restrict__ ws) {
  const int i = blockIdx.x * 256 + threadIdx.x;  // float4 index
  const int e = i * 4;
  const int d = e & 63;
  const int h = (e >> 6) & 7;
  const int b = e >> 22;  // / (8192*8*64)
  float4 o = *(float4*)(out + e);
  const float4 dn = *(const float4*)(ws + WS_DENOM + ((b * HH + h) * DD + d));
  o.x = o.x / (dn.x + 1e-8f) / 3.0f;
  o.y = o.y / (dn.y + 1e-8f) / 3.0f;
  o.z = o.z / (dn.z + 1e-8f) / 3.0f;
  o.w = o.w / (dn.w + 1e-8f) / 3.0f;
  *(float4*)(out + e) = o;
}

extern "C" void kernel_launch(void* const* d_in, const int* in_sizes, int n_in,
                              void* d_out, int out_size, void* d_ws,
                              size_t ws_size, hipStream_t stream) {
  (void)in_sizes; (void)n_in; (void)out_size; (void)ws_size;
  const float* q = (const float*)d_in[0];
  const float* k = (const float*)d_in[1];
  const float* v = (const float*)d_in[2];
  float* out = (float*)d_out;
  float* ws  = (float*)d_ws;

  const dim3 g1(NN / 32, BB);
  dil_pass1<3, 0, 1, 0, WS_P0><<<g1, 256, 0, stream>>>(q, k, v, out, ws);
  dil_pass1<3, 3, 2, 1, WS_P1><<<g1, 256, 0, stream>>>(q, k, v, out, ws);
  dil_pass1<2, 6, 4, 2, WS_P2><<<g1, 256, 0, stream>>>(q, k, v, out, ws);
  dil_reduce<<<(BB * HH * DD + 255) / 256, 256, 0, stream>>>(ws);
  dil_pass2<<<(BB * NN * HH * DD / 4) / 256, 256, 0, stream>>>(out, ws);
}